// SparseConvTranspose_73495480369569
// MI455X (gfx1250) — compile-verified
//
#include <hip/hip_runtime.h>

// ---------------------------------------------------------------------------
// Sparse conv (gather-GEMM-scatter) for MI455X / gfx1250, f32 WMMA path.
//   feats  : [N_IN, 64]  f32
//   weight : [27, 64, 32] f32
//   in_map : [27, N_IN]  i64
//   out_map: [27, N_IN]  i64
//   out    : [N_OUT, 32] f32 (scatter-add)
// One wave32 = one 16-row tile: C[16x32] += A[16x64] * W_k[64x32]
// via V_WMMA_F32_16X16X4_F32 (16 K-steps x 2 N-tiles).
// ---------------------------------------------------------------------------

typedef __attribute__((ext_vector_type(2))) float v2f;
typedef __attribute__((ext_vector_type(8))) float v8f;

#define K_OFF   27
#define C_IN    64
#define C_OUT   32
#define N_IN    100000
#define N_OUT   400000
#define TILES_M (N_IN / 16)                 // 6250 (exact)
#define WPB     8                           // waves per block (256 threads)
#define CHUNKS  ((TILES_M + WPB - 1) / WPB) // 782

__global__ __launch_bounds__(256) void spconv_zero_kernel(float4* __restrict__ out, int n4) {
    int i = blockIdx.x * blockDim.x + threadIdx.x;
    if (i < n4) out[i] = make_float4(0.f, 0.f, 0.f, 0.f);
}

__global__ __launch_bounds__(256) void spconv_wmma_kernel(
    const float* __restrict__ feats,
    const float* __restrict__ weight,
    const long long* __restrict__ in_map,
    const long long* __restrict__ out_map,
    float* __restrict__ out)
{
    __shared__ __align__(16) float wlds[C_IN * C_OUT];   // 8 KB weight tile

    const int tid   = threadIdx.x;
    const int k     = blockIdx.x / CHUNKS;   // kernel offset, uniform per block
    const int chunk = blockIdx.x % CHUNKS;

    // Stage weight[k] (64x32 f32, row-major K x N) into LDS: 512 float4s.
    {
        const float4* wg = (const float4*)(weight + (size_t)k * (C_IN * C_OUT));
        float4* wl = (float4*)wlds;
        wl[tid]       = wg[tid];
        wl[tid + 256] = wg[tid + 256];
    }
    __syncthreads();

    const int wave = tid >> 5;
    const int lane = tid & 31;
    const int half = lane >> 4;   // 0: lanes 0-15, 1: lanes 16-31
    const int nrow = lane & 15;

    const int tm = chunk * WPB + wave;       // uniform within the wave
    if (tm >= TILES_M) return;               // wave-uniform: EXEC stays all-1s
    const int m0 = tm * 16;

    const long long* imap = in_map  + (size_t)k * N_IN + m0;
    const long long* omap = out_map + (size_t)k * N_IN + m0;

    // A-fragment addressing (ISA 7.12.2, 32-bit A 16x4):
    //   lane L (row = L&15): VGPR0 = A[row][4kk + 2*half], VGPR1 = A[row][4kk + 2*half + 1]
    // -> per lane an 8-byte contiguous load from the gathered row.
    const long long inrow = imap[nrow];
    const float* arow = feats + (size_t)inrow * C_IN + 2 * half;

    v8f acc0 = {};   // N-tile 0 (cols 0..15)
    v8f acc1 = {};   // N-tile 1 (cols 16..31)

    #pragma unroll
    for (int kk = 0; kk < 16; ++kk) {
        v2f a = *(const v2f*)(arow + 4 * kk);

        // B-fragment (4x16 per step): VGPR0 = B[K=2*half][N=nrow], VGPR1 = B[K=2*half+1][N=nrow]
        const int krow = 4 * kk + 2 * half;
        v2f b0, b1;
        b0.x = wlds[(krow    ) * C_OUT + nrow];
        b0.y = wlds[(krow + 1) * C_OUT + nrow];
        b1.x = wlds[(krow    ) * C_OUT + 16 + nrow];
        b1.y = wlds[(krow + 1) * C_OUT + 16 + nrow];

        // 8-arg pattern: (neg_a, A, neg_b, B, c_mod, C, reuse_a, reuse_b)
        acc0 = __builtin_amdgcn_wmma_f32_16x16x4_f32(
                   false, a, false, b0, (short)0, acc0, false, false);
        acc1 = __builtin_amdgcn_wmma_f32_16x16x4_f32(
                   false, a, false, b1, (short)0, acc1, false, false);
    }

    // Scatter-add. D layout (32-bit C/D 16x16): VGPR i, lane L holds
    // D[M = (L>>4)*8 + i][N = L&15].
    #pragma unroll
    for (int i = 0; i < 8; ++i) {
        const int M = half * 8 + i;
        const long long orow = omap[M];
        float* obase = out + (size_t)orow * C_OUT;
        atomicAdd(obase + nrow,      acc0[i]);   // global_atomic_add_f32 (no return)
        atomicAdd(obase + 16 + nrow, acc1[i]);
    }
}

extern "C" void kernel_launch(void* const* d_in, const int* in_sizes, int n_in,
                              void* d_out, int out_size, void* d_ws, size_t ws_size,
                              hipStream_t stream) {
    const float*     feats   = (const float*)d_in[0];
    const float*     weight  = (const float*)d_in[1];
    const long long* in_map  = (const long long*)d_in[2];
    const long long* out_map = (const long long*)d_in[3];
    float*           out     = (float*)d_out;

    // Zero the output (harness poisons it; we accumulate atomically).
    const int n4 = (N_OUT * C_OUT) / 4;
    spconv_zero_kernel<<<(n4 + 255) / 256, 256, 0, stream>>>((float4*)out, n4);

    // 27 offsets x 782 chunks of 8 wave-tiles.
    spconv_wmma_kernel<<<dim3(K_OFF * CHUNKS), 256, 0, stream>>>(
        feats, weight, in_map, out_map, out);
}